// CosineAttention_58093727645857
// MI455X (gfx1250) — compile-verified
//
#include <hip/hip_runtime.h>
#include <math.h>

// ---- problem constants (match reference) ----
#define BB 2
#define HH 8
#define SS 4096
#define DD 64
#define BH (BB*HH)          // 16
#define NS 16               // key-dim splits per head
#define CH (SS/NS)          // 256 keys per phase-1 block
#define PSTR 4352           // floats per workspace slot (4096 M + 64 ksum + 64 s + 1 cnt, padded)
#define FINBASE (BH*NS*PSTR) // start of reduced slots
#define LROW 68             // padded LDS row stride (floats): 16B aligned, conflict-free

typedef __attribute__((ext_vector_type(2))) float v2f;
typedef __attribute__((ext_vector_type(8))) float v8f;

__device__ __forceinline__ v8f wmma_f32_4(v2f a, v2f b, v8f c) {
    // D = A(16x4, f32) * B(4x16, f32) + C(16x16, f32)
    return __builtin_amdgcn_wmma_f32_16x16x4_f32(
        false, a, false, b, (short)0, c, false, false);
}

// A/B fragment for 16x16x4 f32 WMMA from an LDS tile stored [k][col] with row
// stride `str`. Lanes 0-15: K={k0,k0+1}, lanes 16-31: K={k0+2,k0+3}; col = M (A)
// or N (B) selected by (lane&15) + tile offset.
__device__ __forceinline__ v2f ld_frag(const float* p, int k0, int col, int lane, int str) {
    int ks = k0 + 2 * (lane >> 4);
    v2f r;
    r.x = p[ks * str + col];
    r.y = p[(ks + 1) * str + col];
    return r;
}

// ---------------- Phase 1: per (head, key-chunk) partial M, ksum, s, cnt -----
__global__ __launch_bounds__(256) void cosattn_p1(
    const float* __restrict__ K, const float* __restrict__ V,
    const int* __restrict__ mask, float* __restrict__ ws) {
    __shared__ float kn[CH * LROW];   // normalized+masked keys, [k][d]
    __shared__ float vv[CH * LROW];   // values, [k][d]
    __shared__ float mf[CH];          // mask as float per key row

    const int tid = threadIdx.x;
    const int blk = blockIdx.x;       // = bh*NS + chunk
    const int bh = blk / NS;
    const int chunk = blk % NS;
    const int b = bh / HH;
    const long gbase = ((long)bh * SS + (long)chunk * CH) * DD;

    // coalesced float4 staging of K and V chunks (CH*DD = 16384 floats each)
#pragma unroll
    for (int j = 0; j < 16; ++j) {
        int e4 = tid + j * 256;               // float4 index, 0..4095
        int row = e4 >> 4;                    // (e4*4)/64
        int col = (e4 & 15) * 4;
        float4 kq = *(const float4*)(K + gbase + (long)e4 * 4);
        float4 vq = *(const float4*)(V + gbase + (long)e4 * 4);
        *(float4*)(kn + row * LROW + col) = kq;
        *(float4*)(vv + row * LROW + col) = vq;
    }
    {
        int kidx = chunk * CH + tid;
        mf[tid] = (mask[b * SS + kidx] != 0) ? 1.0f : 0.0f;
    }
    __syncthreads();

    // normalize + fold mask into key row `tid`
    {
        float* rowp = kn + tid * LROW;
        float ssum = 0.0f;
#pragma unroll
        for (int c = 0; c < DD; c += 4) {
            float4 x = *(float4*)(rowp + c);
            ssum += x.x * x.x + x.y * x.y + x.z * x.z + x.w * x.w;
        }
        float scale = mf[tid] / fmaxf(sqrtf(ssum), 1e-12f);
#pragma unroll
        for (int c = 0; c < DD; c += 4) {
            float4 x = *(float4*)(rowp + c);
            x.x *= scale; x.y *= scale; x.z *= scale; x.w *= scale;
            *(float4*)(rowp + c) = x;
        }
    }
    __syncthreads();

    const long pbase = (long)blk * PSTR;

    // partial ksum (threads 0-63), s (64-127), cnt (128)
    if (tid < 64) {
        float a = 0.0f;
        for (int r = 0; r < CH; ++r) a += kn[r * LROW + tid];
        ws[pbase + 4096 + tid] = a;
    } else if (tid < 128) {
        int c = tid - 64;
        float a = 0.0f;
        for (int r = 0; r < CH; ++r) a += mf[r] * vv[r * LROW + c];
        ws[pbase + 4160 + c] = a;
    } else if (tid == 128) {
        float a = 0.0f;
        for (int r = 0; r < CH; ++r) a += mf[r];
        ws[pbase + 4224] = a;
    }
    __syncthreads();   // control flow uniform again -> EXEC all-1s for WMMA

    // partial M = K_n^T * V over this chunk; 8 waves x 2 (16x16) tiles = 64x64
    const int lane = tid & 31;
    const int wv = tid >> 5;
    const int ti = wv >> 1;            // d1 tile (rows of M)
    const int tj = (wv & 1) * 2;       // first d2 tile (cols of M)
    const int colA = ti * 16 + (lane & 15);
    const int colB0 = tj * 16 + (lane & 15);
    v8f acc0 = {};
    v8f acc1 = {};
    for (int k0 = 0; k0 < CH; k0 += 4) {
        v2f a  = ld_frag(kn, k0, colA, lane, LROW);
        v2f b0 = ld_frag(vv, k0, colB0, lane, LROW);
        v2f b1 = ld_frag(vv, k0, colB0 + 16, lane, LROW);
        acc0 = wmma_f32_4(a, b0, acc0);
        acc1 = wmma_f32_4(a, b1, acc1);
    }
    // D layout: lane L, vgpr r -> row = r + 8*(L>>4), col = L&15 (within tile)
    const int rbase = ti * 16 + 8 * (lane >> 4);
    const int c0 = tj * 16 + (lane & 15);
#pragma unroll
    for (int r = 0; r < 8; ++r) {
        ws[pbase + (rbase + r) * 64 + c0] = acc0[r];
        ws[pbase + (rbase + r) * 64 + c0 + 16] = acc1[r];
    }
}

// ---------------- Phase 1b: deterministic reduction over NS chunk partials ---
__global__ __launch_bounds__(256) void cosattn_reduce(float* __restrict__ ws) {
    const int slot = blockIdx.x / 17;                      // bh
    const int e = (blockIdx.x % 17) * 256 + threadIdx.x;   // element in slot
    if (e >= 4225) return;
    const long pb = (long)slot * NS * PSTR;
    float a = 0.0f;
#pragma unroll
    for (int c = 0; c < NS; ++c) a += ws[pb + (long)c * PSTR + e];
    ws[FINBASE + (long)slot * PSTR + e] = a;
}

// ---------------- Phase 2: out = (q_n * M + s) / (q_n . ksum + cnt) ----------
__global__ __launch_bounds__(256) void cosattn_p2(
    const float* __restrict__ Q, const float* __restrict__ ws,
    float* __restrict__ out) {
    __shared__ float fin[PSTR];            // [0,4096) M, [4096,4160) ksum, [4160,4224) s, [4224] cnt
    __shared__ float qs[8 * 16 * LROW];    // per-wave 16x64 q tiles (padded)
    __shared__ float scale_s[128];
    __shared__ float denom_s[128];

    const int tid = threadIdx.x;
    const int lane = tid & 31;
    const int wv = tid >> 5;
    const int bh = blockIdx.x >> 5;         // 16 heads
    const int rg = blockIdx.x & 31;         // 32 row groups of 128
    const int q0 = rg * 128;

    const float* finp = ws + FINBASE + (long)bh * PSTR;
    for (int e = tid; e < 4225; e += 256) fin[e] = finp[e];

    const long qrowbase = (long)bh * SS + q0 + wv * 16;   // first q row of this wave
    float* myq = qs + wv * 16 * LROW;
#pragma unroll
    for (int j = 0; j < 16; ++j) {
        float2 x = *(const float2*)(Q + (qrowbase + j) * DD + lane * 2);
        *(float2*)(myq + j * LROW + lane * 2) = x;
    }
    __syncthreads();

    // per-row 1/norm and denominator (2 lanes per row, halves of D)
    const int r = lane & 15;
    const int half = lane >> 4;
    {
        float ssum = 0.0f, kd = 0.0f;
#pragma unroll
        for (int c32 = 0; c32 < 32; ++c32) {
            int c = half * 32 + c32;
            float v = myq[r * LROW + c];
            ssum += v * v;
            kd += v * fin[4096 + c];
        }
        ssum += __shfl_xor(ssum, 16, 32);
        kd += __shfl_xor(kd, 16, 32);
        float sc = 1.0f / fmaxf(sqrtf(ssum), 1e-12f);
        if (lane < 16) {
            scale_s[wv * 16 + lane] = sc;
            denom_s[wv * 16 + lane] = sc * kd + fin[4224];
        }
    }
    __syncthreads();

    // 16 rows x 64 cols = 4 WMMA tiles; C initialized with s (bias) per column
    v8f acc[4];
#pragma unroll
    for (int nt = 0; nt < 4; ++nt) {
        float sv = fin[4160 + nt * 16 + r];
        v8f ci;
#pragma unroll
        for (int i = 0; i < 8; ++i) ci[i] = sv;
        acc[nt] = ci;
    }
    const float scr = scale_s[wv * 16 + r];   // scale for this lane's A-frag row
    for (int k0 = 0; k0 < DD; k0 += 4) {
        int ks = k0 + 2 * half;
        v2f a;
        a.x = myq[r * LROW + ks] * scr;
        a.y = myq[r * LROW + ks + 1] * scr;
#pragma unroll
        for (int nt = 0; nt < 4; ++nt) {
            int n = nt * 16 + r;
            v2f bfr;
            bfr.x = fin[ks * 64 + n];
            bfr.y = fin[(ks + 1) * 64 + n];
            acc[nt] = wmma_f32_4(a, bfr, acc[nt]);
        }
    }

    // store: lane L, vgpr rr -> row = rr + 8*half, col = nt*16 + (L&15)
#pragma unroll
    for (int nt = 0; nt < 4; ++nt) {
        int col = nt * 16 + r;
#pragma unroll
        for (int rr = 0; rr < 8; ++rr) {
            int rowT = rr + 8 * half;
            float dv = denom_s[wv * 16 + rowT];
            out[(qrowbase + rowT) * DD + col] = acc[nt][rr] / dv;
        }
    }
}

extern "C" void kernel_launch(void* const* d_in, const int* in_sizes, int n_in,
                              void* d_out, int out_size, void* d_ws, size_t ws_size,
                              hipStream_t stream) {
    const float* Q = (const float*)d_in[0];
    const float* K = (const float*)d_in[1];
    const float* V = (const float*)d_in[2];
    const int* M = (const int*)d_in[3];
    float* out = (float*)d_out;
    float* ws = (float*)d_ws;

    cosattn_p1<<<dim3(BH * NS), dim3(256), 0, stream>>>(K, V, M, ws);
    cosattn_reduce<<<dim3(BH * 17), dim3(256), 0, stream>>>(ws);
    cosattn_p2<<<dim3(BH * 32), dim3(256), 0, stream>>>(Q, ws, out);
}